// FuzzyDNF_50233937494029
// MI455X (gfx1250) — compile-verified
//
#include <hip/hip_runtime.h>

// FuzzyDNF: B=128, IN_F=1024, HID_F=1024, OUT_F=512, all float32.
#define B_    128
#define INF_  1024
#define HIDF_ 1024
#define OUTF_ 512

typedef __attribute__((ext_vector_type(2))) float v2f;
typedef __attribute__((ext_vector_type(8))) float v8f;

#if __has_builtin(__builtin_amdgcn_wmma_f32_16x16x4_f32)
#define HAVE_WMMA4 1
#else
#define HAVE_WMMA4 0
#endif

// Workspace layout (float offsets)
#define WS_A   0u
#define WS_C   (1024u * 1024u)
#define WS_W2S (2u * 1024u * 1024u)
#define WS_H   (WS_W2S + 1024u * 512u)   // 128*1024 floats

__device__ __forceinline__ float sigmoidf_(float v) {
    return 1.0f / (1.0f + __expf(-v));
}

// ---------------------------------------------------------------------------
// Kernel 1: elementwise precompute.
//   A[i,j] = 1 - sw*ss          (sw = sigmoid(w1), ss = sigmoid(s1))
//   C[i,j] = sw*(1 - 2*ss)      so term = A - x*C
//   W2S    = sigmoid(w2)
// ---------------------------------------------------------------------------
__global__ __launch_bounds__(256) void fdnf_prep_kernel(
    const float* __restrict__ w1, const float* __restrict__ s1,
    const float* __restrict__ w2, float* __restrict__ ws) {
    unsigned idx = blockIdx.x * 256u + threadIdx.x;
    if (idx < INF_ * HIDF_) {
        float sw = sigmoidf_(w1[idx]);
        float ss = sigmoidf_(s1[idx]);
        ws[WS_A + idx] = __builtin_fmaf(-sw, ss, 1.0f);
        ws[WS_C + idx] = sw * __builtin_fmaf(-2.0f, ss, 1.0f);
    }
    if (idx < HIDF_ * OUTF_) {
        ws[WS_W2S + idx] = sigmoidf_(w2[idx]);
    }
}

// ---------------------------------------------------------------------------
// Kernel 2: layer 1, exact fp32 product reduction.
//   h[b,j] = prod_i (A[i,j] - x[b,i]*C[i,j])
// Block: 256 threads = 256 consecutive j columns; 8 batch rows per block,
// x-tile (8x1024 = 32KB) staged in LDS (broadcast reads). A/C loads are
// fully coalesced and L2-resident. Grid: 4 j-blocks x 16 b-blocks.
// ---------------------------------------------------------------------------
__global__ __launch_bounds__(256) void fdnf_layer1_kernel(
    const float* __restrict__ x, float* __restrict__ ws) {
    const float* Amat = ws + WS_A;
    const float* Cmat = ws + WS_C;
    __shared__ float xs[8 * INF_];

    const unsigned tid = threadIdx.x;
    const unsigned j   = blockIdx.x * 256u + tid;
    const unsigned b0  = blockIdx.y * 8u;

    for (unsigned t = tid; t < 8u * INF_; t += 256u) {
        unsigned r = t >> 10, i = t & (INF_ - 1u);
        xs[t] = x[(b0 + r) * INF_ + i];
    }
    __syncthreads();

    float prod[8];
#pragma unroll
    for (int r = 0; r < 8; ++r) prod[r] = 1.0f;

    for (unsigned i = 0; i < INF_; ++i) {
        float a = Amat[i * HIDF_ + j];
        float c = Cmat[i * HIDF_ + j];
#pragma unroll
        for (int r = 0; r < 8; ++r) {
            prod[r] *= __builtin_fmaf(-xs[r * INF_ + i], c, a);
        }
    }
#pragma unroll
    for (int r = 0; r < 8; ++r) {
        ws[WS_H + (b0 + r) * HIDF_ + j] = prod[r];
    }
}

// ---------------------------------------------------------------------------
// Kernel 3: layer 2 via log-space Taylor GEMM on WMMA.
//   out[b,k] = 1 - exp( sum_j log(1 - w2s[j,k]*h[b,j]) )
//            = 1 - exp( -sum_n (1/n) * sum_j h^n * w2s^n )   (|z|<1)
// Each Taylor term is a 128x512x1024 GEMM; elementwise powers preserve the
// WMMA fragment layout, so we power the fragments in registers and issue
// V_WMMA_F32_16X16X4_F32 per (K-step, n). One wave per 16x16 output tile.
// A-frag (16x4 f32): lanes 0-15 hold M=0..15 K={0,1}; lanes 16-31 K={2,3}.
// B-frag (4x16 f32): mirrored; C/D: 8 VGPRs, lane<16 -> M=v, lane>=16 -> M=v+8.
// ---------------------------------------------------------------------------
#define NTAYLOR 8

__global__ __launch_bounds__(32) void fdnf_layer2_kernel(
    const float* __restrict__ ws, float* __restrict__ out) {
    const float* H   = ws + WS_H;
    const float* W2S = ws + WS_W2S;

    const unsigned lane = threadIdx.x;
    const unsigned half = lane >> 4;
    const unsigned l16  = lane & 15u;
    const unsigned n0   = blockIdx.x * 16u;   // output-feature tile
    const unsigned m0   = blockIdx.y * 16u;   // batch tile

#if HAVE_WMMA4
    v8f acc = {};
    for (unsigned j = 0; j < HIDF_; j += 4) {
        const unsigned ka = j + 2u * half;
        v2f a, b;
        a[0] = H[(m0 + l16) * HIDF_ + ka];
        a[1] = H[(m0 + l16) * HIDF_ + ka + 1u];
        b[0] = W2S[ka * OUTF_ + (n0 + l16)];
        b[1] = W2S[(ka + 1u) * OUTF_ + (n0 + l16)];

        v2f ap = a, bp = b;
#pragma unroll
        for (int n = 1; n <= NTAYLOR; ++n) {
            const float inv = 1.0f / (float)n;
            v2f as;
            as[0] = ap[0] * inv;
            as[1] = ap[1] * inv;
            // (neg_a, A, neg_b, B, c_mod, C, reuse_a, reuse_b)
            acc = __builtin_amdgcn_wmma_f32_16x16x4_f32(
                false, as, false, bp, (short)0, acc, false, false);
            ap[0] *= a[0]; ap[1] *= a[1];
            bp[0] *= b[0]; bp[1] *= b[1];
        }
    }
#pragma unroll
    for (int v = 0; v < 8; ++v) {
        const unsigned m = m0 + (unsigned)v + 8u * half;
        out[m * OUTF_ + (n0 + l16)] = 1.0f - __expf(-acc[v]);
    }
#else
    // Exact scalar fallback (same output mapping as the WMMA store).
    float acc[8];
#pragma unroll
    for (int v = 0; v < 8; ++v) acc[v] = 0.0f;
    for (unsigned j = 0; j < HIDF_; ++j) {
        const float w2v = W2S[j * OUTF_ + (n0 + l16)];
#pragma unroll
        for (int v = 0; v < 8; ++v) {
            const unsigned m = m0 + (unsigned)v + 8u * half;
            acc[v] += log1pf(-w2v * H[m * HIDF_ + j]);
        }
    }
#pragma unroll
    for (int v = 0; v < 8; ++v) {
        const unsigned m = m0 + (unsigned)v + 8u * half;
        out[m * OUTF_ + (n0 + l16)] = 1.0f - __expf(acc[v]);
    }
#endif
}

// ---------------------------------------------------------------------------
extern "C" void kernel_launch(void* const* d_in, const int* in_sizes, int n_in,
                              void* d_out, int out_size, void* d_ws, size_t ws_size,
                              hipStream_t stream) {
    const float* x  = (const float*)d_in[0];
    const float* w1 = (const float*)d_in[1];
    const float* s1 = (const float*)d_in[2];
    const float* w2 = (const float*)d_in[3];
    float* ws  = (float*)d_ws;
    float* out = (float*)d_out;

    (void)in_sizes; (void)n_in; (void)out_size; (void)ws_size;

    fdnf_prep_kernel<<<(INF_ * HIDF_ + 255) / 256, 256, 0, stream>>>(w1, s1, w2, ws);
    fdnf_layer1_kernel<<<dim3(HIDF_ / 256, B_ / 8), 256, 0, stream>>>(x, ws);
    fdnf_layer2_kernel<<<dim3(OUTF_ / 16, B_ / 16), 32, 0, stream>>>(ws, out);
}